// SimpleUpscaler_90769838834203
// MI455X (gfx1250) — compile-verified
//
#include <hip/hip_runtime.h>
#include <cstdint>

#define VOCAB_SZ 100000
#define EMBED    256
#define KIDS     8

// One block per (b,s) position; thread t owns embedding dim e = t.
// Gathers W[e*V + id[k]] for k=0..7 via gfx1250 async loads into LDS,
// waits once on ASYNCcnt, then sums with the id!=0 mask and adds bias.
__global__ __launch_bounds__(EMBED)
void embedbag_async_kernel(const int* __restrict__ ids,
                           const float* __restrict__ W,
                           const float* __restrict__ bias,
                           float* __restrict__ out)
{
    __shared__ float tile[KIDS * EMBED];   // 8 KB staging for async gathers

    const int bs = blockIdx.x;             // position index in [0, 3200)
    const int e  = threadIdx.x;            // embedding dim in [0, 256)

    // 8 ids for this position (uniform across the block -> scalarizes)
    const int* idp = ids + (size_t)bs * KIDS;
    int id[KIDS];
#pragma unroll
    for (int k = 0; k < KIDS; ++k) id[k] = idp[k];

    const float* wrow = W + (size_t)e * VOCAB_SZ;

    // Fire-and-forget 8 async gathers: memory -> LDS, tracked by ASYNCcnt.
    // Unconditional (W[e*V+0] is valid memory); mask applied at accumulate
    // so EXEC stays all-ones and no divergence at issue.
#pragma unroll
    for (int k = 0; k < KIDS; ++k) {
        unsigned long long gaddr =
            (unsigned long long)(const void*)(wrow + (unsigned)id[k]);
        // generic pointer low 32 bits == wave-relative LDS byte offset
        unsigned lds_off = (unsigned)(size_t)&tile[k * EMBED + e];
        asm volatile("global_load_async_to_lds_b32 %0, %1, off"
                     :
                     : "v"(lds_off), "v"(gaddr)
                     : "memory");
    }

    // Single wait for all 8 outstanding async transfers of this wave.
    asm volatile("s_wait_asynccnt 0x0" ::: "memory");

    float acc = bias[e];
#pragma unroll
    for (int k = 0; k < KIDS; ++k)
        acc += (id[k] != 0) ? tile[k * EMBED + e] : 0.0f;

    out[(size_t)bs * EMBED + e] = acc;
}

extern "C" void kernel_launch(void* const* d_in, const int* in_sizes, int n_in,
                              void* d_out, int out_size, void* d_ws, size_t ws_size,
                              hipStream_t stream) {
    const int*   ids  = (const int*)d_in[0];   // content_input [16,200,8]
    const float* W    = (const float*)d_in[1]; // [256, 100000]
    const float* bias = (const float*)d_in[2]; // [256]
    float*       out  = (float*)d_out;         // [16,200,256]

    const int npos = in_sizes[0] / KIDS;       // 16*200 = 3200 positions
    embedbag_async_kernel<<<npos, EMBED, 0, stream>>>(ids, W, bias, out);
    (void)n_in; (void)out_size; (void)d_ws; (void)ws_size;
}